// Edge_to_Edge_Conv2d_79164837200687
// MI455X (gfx1250) — compile-verified
//
#include <hip/hip_runtime.h>
#include <hip/hip_bf16.h>

// ---------------------------------------------------------------------------
// Edge_to_Edge_Conv2d  (B=32, K=512, C=32)
//
// s[b,o] = sum_t (x_b @ w_col[o] + b_col[o])_t * (x_b^T @ w_row[o] + b_row[o])_t
// out[b,o,:,:] = s[b,o]  (broadcast to 512x512)
//
// Cost model (MI455X): compute = ~1.07 GFLOP fp32 (negligible);
// output = 1 GiB of stores -> ~46us at 23.3 TB/s. Kernel B (broadcast) is the
// whole runtime; kernel A uses V_WMMA_F32_16X16X4_F32 for exact fp32 GEMMs.
// ---------------------------------------------------------------------------

typedef __attribute__((ext_vector_type(2))) float v2f;
typedef __attribute__((ext_vector_type(4))) float v4f;
typedef __attribute__((ext_vector_type(8))) float v8f;

#define KDIM 512
#define CDIM 32
#define BDIM 32

// WMMA layout assumptions (ISA 7.12.2, f32 16x16x4):
//   A (16x4):  lane l: M = l%16, VGPR0 holds K=2*(l/16), VGPR1 holds K=2*(l/16)+1
//   B (4x16):  lane l: N = l%16, VGPR0 holds K=2*(l/16), VGPR1 holds K=2*(l/16)+1
//   C/D (16x16): VGPR r, lane l: M = r + 8*(l/16), N = l%16
__global__ __launch_bounds__(256) void e2e_compute_s(
    const float* __restrict__ x,      // (B, 1, K, K)
    const float* __restrict__ w_col,  // (C, K)
    const float* __restrict__ b_col,  // (C,)
    const float* __restrict__ w_row,  // (C, K)
    const float* __restrict__ b_row,  // (C,)
    float* __restrict__ s_buf)        // (B, C)  pre-zeroed
{
    const int b    = blockIdx.x;          // batch
    const int ot   = blockIdx.y;          // o-tile: 0..1 (16 channels each)
    const int tid  = threadIdx.x;
    const int wave = tid >> 5;            // 8 waves
    const int lane = tid & 31;
    const int lh   = lane >> 4;           // lane half
    const int ln   = lane & 15;

    // per-wave 16x16 exchange patch, padded to 17 to kill bank conflicts
    __shared__ float patch[8][16][17];

    const float* xb = x + (size_t)b * KDIM * KDIM;
    const int o = ot * 16 + ln;           // this lane's channel (same for all 8 acc VGPRs)
    const float bc = b_col[o];
    const float br = b_row[o];

    // Per-lane bases (k advances in the loop):
    const float* wcb = w_col + (size_t)o * KDIM + 2 * lh;  // B for Col GEMM
    const float* wra = w_row + (size_t)o * KDIM + 2 * lh;  // A for Row GEMM
    float p = 0.0f;  // partial of s[b, o] over this lane's (t) coverage

    for (int q = 0; q < 4; ++q) {
        const int tt = wave + 8 * q;      // t-tile 0..31 (16 t values)
        // A for Col GEMM: x[i = tt*16+ln, k + 2*lh .. +1]  (contiguous float2)
        const float* xa  = xb + (size_t)(tt * 16 + ln) * KDIM + 2 * lh;
        // B for Row GEMM: x[k + 2*lh (+1 row), j = tt*16+ln]
        const float* xrb = xb + (size_t)(2 * lh) * KDIM + (tt * 16 + ln);

        v8f colAcc = {};  // D[i(t), o]
        v8f rowAcc = {};  // D[o, j(t)]
        #pragma unroll 4
        for (int k0 = 0; k0 < KDIM; k0 += 4) {
            v2f a  = *(const v2f*)(xa  + k0);
            v2f wb = *(const v2f*)(wcb + k0);
            colAcc = __builtin_amdgcn_wmma_f32_16x16x4_f32(
                false, a, false, wb, (short)0, colAcc, false, false);

            v2f wa = *(const v2f*)(wra + k0);
            v2f xv;
            xv.x = xrb[(size_t)k0 * KDIM];
            xv.y = xrb[(size_t)k0 * KDIM + KDIM];
            rowAcc = __builtin_amdgcn_wmma_f32_16x16x4_f32(
                false, wa, false, xv, (short)0, rowAcc, false, false);
        }

        // Transpose Row tile through LDS so it aligns with Col tile layout.
        // Row D: (o_local = r + 8*lh, t_local = ln) -> patch[wave][o][t]
        union { v8f v; float f[8]; } ru, cu;
        ru.v = rowAcc; cu.v = colAcc;
        #pragma unroll
        for (int r = 0; r < 8; ++r)
            patch[wave][r + 8 * lh][ln] = ru.f[r];
        // Col D element r: (t_local = r + 8*lh, o_local = ln)
        // matching Row value: patch[wave][ln][r + 8*lh]  (same wave: LDS in-order)
        #pragma unroll
        for (int r = 0; r < 8; ++r) {
            float rowv = patch[wave][ln][r + 8 * lh];
            p += (cu.f[r] + bc) * (rowv + br);
        }
    }

    // lanes l and l+16 hold the same o, complementary t halves
    p += __shfl_xor(p, 16, 32);
    if (lh == 0)
        atomicAdd(&s_buf[b * CDIM + o], p);
}

// Broadcast: out[b][o][.][.] = s[b,o]. 1 GiB of pure streaming stores.
// grid = (1024 pairs, 16 chunks), 256 threads; 16 x b128 NT stores per thread.
__global__ __launch_bounds__(256) void e2e_broadcast(
    const float* __restrict__ s_buf, v4f* __restrict__ out)
{
    const int pair  = blockIdx.x;                 // b*32 + o
    const int chunk = blockIdx.y;                 // 0..15
    const float s = s_buf[pair];
    v4f v; v.x = s; v.y = s; v.z = s; v.w = s;
    // per pair: 512*512 floats = 65536 v4f; per chunk: 4096 v4f; per thread: 16
    size_t base = (size_t)pair * 65536 + (size_t)chunk * 4096 + threadIdx.x;
    #pragma unroll
    for (int j = 0; j < 16; ++j)
        __builtin_nontemporal_store(v, &out[base + (size_t)j * 256]);
}

extern "C" void kernel_launch(void* const* d_in, const int* in_sizes, int n_in,
                              void* d_out, int out_size, void* d_ws, size_t ws_size,
                              hipStream_t stream) {
    (void)in_sizes; (void)n_in; (void)out_size; (void)ws_size;
    const float* x     = (const float*)d_in[0];
    const float* w_col = (const float*)d_in[1];
    const float* b_col = (const float*)d_in[2];
    const float* w_row = (const float*)d_in[3];
    const float* b_row = (const float*)d_in[4];
    float* s_buf = (float*)d_ws;                  // 32*32 floats

    hipMemsetAsync(s_buf, 0, BDIM * CDIM * sizeof(float), stream);

    dim3 gA(BDIM, 2);
    e2e_compute_s<<<gA, 256, 0, stream>>>(x, w_col, b_col, w_row, b_row, s_buf);

    dim3 gB(BDIM * CDIM, 16);
    e2e_broadcast<<<gB, 256, 0, stream>>>(s_buf, (v4f*)d_out);
}